// WassersteinLoss_16896401343148
// MI455X (gfx1250) — compile-verified
//
#include <hip/hip_runtime.h>
#include <stdint.h>

// ---------------------------------------------------------------------------
// WassersteinLoss for MI455X (gfx1250, wave32)
//   d = y_pred - y_true ; c = cumsum(d, axis=T) ; out = scale * mean_b sum |c|*(T-t)
// Memory-bound: 268 MB of reads -> ~11.5 us at 23.3 TB/s.
// One block per (b,c) row; TDM double-buffered LDS staging; WMMA (L * X with
// lower-triangular ones) performs the 16-segment scans; carries via bpermute
// lane scan + LDS wave scan + uniform running block carry.
// ---------------------------------------------------------------------------

#define TLEN   32768
#define NROWS  1024          // B*C
#define BDIM   256           // 8 waves
#define WPB    8
#define CHUNK  256           // elements per wave per tile (16x16 matrix)
#define TILE   (WPB * CHUNK) // 2048 floats per tile
#define NTILES (TLEN / TILE) // 16

typedef __attribute__((ext_vector_type(2))) float        v2f;
typedef __attribute__((ext_vector_type(8))) float        v8f;
typedef __attribute__((ext_vector_type(4))) unsigned int v4u;
typedef __attribute__((ext_vector_type(4))) int          v4i;
typedef __attribute__((ext_vector_type(8))) int          v8i;

#if __has_builtin(__builtin_amdgcn_tensor_load_to_lds) && __has_builtin(__builtin_amdgcn_s_wait_tensorcnt)
#define USE_TDM 1
#else
#define USE_TDM 0
#endif

#if __has_builtin(__builtin_amdgcn_wmma_f32_16x16x4_f32)
#define USE_WMMA 1
#else
#define USE_WMMA 0
#endif

__device__ __forceinline__ float bpf(float v, int srcLane) {
    return __int_as_float(__builtin_amdgcn_ds_bpermute(srcLane << 2, __float_as_int(v)));
}

#if USE_TDM
// Build a 1-D TDM descriptor (ISA cdna5 ch8): count=1, type=2("image"),
// data_size=4B, tensor_dim0 = tile_dim0 = TILE elements, strides unused.
// clang-23 / therock signature: (v4u g0, v8i g1, v4i g2, v4i g3, v8i, i32 cpol)
__device__ __forceinline__ void tdm_load_tile(const float* gptr, unsigned int lds_byte_off) {
    unsigned long long ga = (unsigned long long)(uintptr_t)gptr;
    v4u g0;
    g0.x = 1u;                                                   // count=1, user mode
    g0.y = lds_byte_off;                                         // lds_addr [63:32]
    g0.z = (unsigned int)ga;                                     // global_addr lo
    g0.w = (unsigned int)((ga >> 32) & 0x01FFFFFFu) | 0x80000000u; // ga hi | type=2
    v8i g1;
    g1[0] = (int)(2u << 16);                                     // data_size = 4B
    g1[1] = (int)((unsigned int)TILE << 16);                     // tensor_dim0[15:0]
    g1[2] = (int)((TILE >> 16) | (1u << 16));                    // tensor_dim0[31:16] | tensor_dim1 lo = 1
    g1[3] = (int)((unsigned int)TILE << 16);                     // tensor_dim1 hi=0 | tile_dim0
    g1[4] = 0; g1[5] = 0; g1[6] = 0; g1[7] = 0;                  // tile_dim1/2 unused, strides 0
    v4i z4 = {0, 0, 0, 0};
    v8i z8 = {0, 0, 0, 0, 0, 0, 0, 0};
    __builtin_amdgcn_tensor_load_to_lds(g0, g1, z4, z4, z8, 0);
}
#endif

__global__ __launch_bounds__(BDIM) void wl_rowscan(const float* __restrict__ yp,
                                                   const float* __restrict__ yt,
                                                   float* __restrict__ rowout) {
    __shared__ float lp[2][TILE];
    __shared__ float lt[2][TILE];
    __shared__ float wsum[WPB];
    __shared__ float red[BDIM];

    const int tid  = threadIdx.x;
    const int lane = tid & 31;
    const int wave = tid >> 5;
    const int col  = lane & 15;      // also row index M for the A fragment
    const int hi   = lane >> 4;
    const int kb   = hi * 2;         // A/B fragment K base per lane half
    const int row  = blockIdx.x;

    const float* gp = yp + (size_t)row * TLEN;
    const float* gt = yt + (size_t)row * TLEN;

    float acc   = 0.f;   // per-thread weighted-abs partial
    float carry = 0.f;   // uniform running prefix carry across tiles

#if USE_TDM
    if (wave == 0) {
        tdm_load_tile(gp, (unsigned int)(uintptr_t)&lp[0][0]);
        tdm_load_tile(gt, (unsigned int)(uintptr_t)&lt[0][0]);
    }
#endif

    for (int tile = 0; tile < NTILES; ++tile) {
        const int buf = tile & 1;
#if USE_TDM
        if (wave == 0) __builtin_amdgcn_s_wait_tensorcnt(0);
        __syncthreads();  // tile data visible; previous buffer fully consumed
        if (wave == 0 && (tile + 1) < NTILES) {
            tdm_load_tile(gp + (size_t)(tile + 1) * TILE, (unsigned int)(uintptr_t)&lp[buf ^ 1][0]);
            tdm_load_tile(gt + (size_t)(tile + 1) * TILE, (unsigned int)(uintptr_t)&lt[buf ^ 1][0]);
        }
#else
        __syncthreads();  // previous iteration done reading lds
        for (int i = tid; i < TILE / 4; i += BDIM) {
            ((float4*)lp[buf])[i] = ((const float4*)(gp + (size_t)tile * TILE))[i];
            ((float4*)lt[buf])[i] = ((const float4*)(gt + (size_t)tile * TILE))[i];
        }
        __syncthreads();
#endif
        const int cbase = wave * CHUNK;
        float chunk_total = 0.f;

#if USE_WMMA
        // P = L * X : X[k][j] = d[cbase + j*16 + k]; four K=4 WMMAs chained.
        v8f c = {0.f, 0.f, 0.f, 0.f, 0.f, 0.f, 0.f, 0.f};
        float seg_excl = 0.f;
        const v2f* p2 = (const v2f*)&lp[buf][cbase + col * 16];
        const v2f* t2 = (const v2f*)&lt[buf][cbase + col * 16];
        #pragma unroll
        for (int i = 0; i < 4; ++i) {
            v2f a;  // A fragment: L[m][4i+kb], L[m][4i+kb+1] (m = col)
            a.x = (col >= 4 * i + kb)     ? 1.f : 0.f;
            a.y = (col >= 4 * i + kb + 1) ? 1.f : 0.f;
            v2f pv = p2[(4 * i + kb) >> 1];
            v2f tv = t2[(4 * i + kb) >> 1];
            v2f b;  // B fragment: X[4i+kb][col], X[4i+kb+1][col]
            b.x = pv.x - tv.x;
            b.y = pv.y - tv.y;
            c = __builtin_amdgcn_wmma_f32_16x16x4_f32(false, a, false, b, (short)0, c,
                                                      false, false);
        }
        // Segment totals = row M=15 of P -> c[7] of lanes 16..31.
        float cs = bpf(c[7], 16 + col);
        float incl = cs;
        #pragma unroll
        for (int d = 1; d < 16; d <<= 1) {
            float up = bpf(incl, lane - d);
            if (col >= d) incl += up;
        }
        seg_excl    = incl - cs;
        chunk_total = bpf(incl, 15);
#else
        // Fallback: per-lane serial scan of 8 contiguous elements + lane scan.
        float v[8];
        float lane_excl = 0.f;
        {
            float run = 0.f;
            #pragma unroll
            for (int k = 0; k < 8; ++k) {
                int idx = cbase + lane * 8 + k;
                run += lp[buf][idx] - lt[buf][idx];
                v[k] = run;
            }
            float incl = run;
            #pragma unroll
            for (int d = 1; d < 32; d <<= 1) {
                float up = bpf(incl, lane - d);
                if (lane >= d) incl += up;
            }
            lane_excl   = incl - run;
            chunk_total = bpf(incl, 31);
        }
#endif
        if (lane == 0) wsum[wave] = chunk_total;
        __syncthreads();
        float wave_excl  = carry;
        float tile_total = 0.f;
        #pragma unroll
        for (int w = 0; w < WPB; ++w) {
            float s = wsum[w];
            if (w < wave) wave_excl += s;
            tile_total += s;
        }
        const int tbase = tile * TILE + cbase;
#if USE_WMMA
        #pragma unroll
        for (int r = 0; r < 8; ++r) {
            int   m   = r + hi * 8;                 // C/D layout: VGPR r, lane half -> row
            int   t   = tbase + col * 16 + m;
            float val = c[r] + seg_excl + wave_excl;
            acc += fabsf(val) * (float)(TLEN - t);
        }
#else
        #pragma unroll
        for (int k = 0; k < 8; ++k) {
            int   t   = tbase + lane * 8 + k;
            float val = v[k] + lane_excl + wave_excl;
            acc += fabsf(val) * (float)(TLEN - t);
        }
#endif
        carry += tile_total;
    }

    // Deterministic block reduction of per-thread partials.
    red[tid] = acc;
    __syncthreads();
    #pragma unroll
    for (int s = BDIM / 2; s > 0; s >>= 1) {
        if (tid < s) red[tid] += red[tid + s];
        __syncthreads();
    }
    if (tid == 0) rowout[row] = red[0];
}

__global__ __launch_bounds__(256) void wl_reduce(const float* __restrict__ rowsums,
                                                 float* __restrict__ out) {
    __shared__ float red[256];
    float s = 0.f;
    for (int i = threadIdx.x; i < NROWS; i += 256) s += rowsums[i];
    red[threadIdx.x] = s;
    __syncthreads();
    #pragma unroll
    for (int k = 128; k > 0; k >>= 1) {
        if ((int)threadIdx.x < k) red[threadIdx.x] += red[threadIdx.x + k];
        __syncthreads();
    }
    if (threadIdx.x == 0) {
        const double TC    = 16.0 * 32768.0;                 // C*T
        const double scale = 2.0 / (TC * (TC + 1.0)) / 64.0; // scaling * mean over B
        out[0] = (float)(scale * (double)red[0]);
    }
}

extern "C" void kernel_launch(void* const* d_in, const int* in_sizes, int n_in,
                              void* d_out, int out_size, void* d_ws, size_t ws_size,
                              hipStream_t stream) {
    const float* yp = (const float*)d_in[0];
    const float* yt = (const float*)d_in[1];
    float* rowsums  = (float*)d_ws;  // NROWS floats of scratch

    wl_rowscan<<<NROWS, BDIM, 0, stream>>>(yp, yt, rowsums);
    wl_reduce<<<1, 256, 0, stream>>>(rowsums, (float*)d_out);
}